// RGCN_21045339750964
// MI455X (gfx1250) — compile-verified
//
#include <hip/hip_runtime.h>
#include <hip/hip_bf16.h>

// Problem constants (from reference)
#define NN 50000
#define EE 600000
#define FF 128
#define HH 128
#define OO 64
#define RR 8

typedef float v2f __attribute__((ext_vector_type(2)));
typedef float v8f __attribute__((ext_vector_type(8)));

// ---------------------------------------------------------------------------
// init min/max buffer: mm[0]=+inf bits (min), mm[1]=0 (max). weights are >= 0
// so uint bit-pattern ordering == float ordering.
// ---------------------------------------------------------------------------
__global__ void init_mm_kernel(unsigned* mm) {
    if (threadIdx.x == 0) { mm[0] = 0x7F800000u; mm[1] = 0u; }
}

// ---------------------------------------------------------------------------
// edge-weight min/max: grid-stride + wave32 shuffle reduce + uint atomics
// ---------------------------------------------------------------------------
__global__ void ew_minmax_kernel(const float* __restrict__ ew, int E,
                                 unsigned* __restrict__ mm) {
    float mn = __builtin_huge_valf();
    float mx = 0.0f;
    for (int i = blockIdx.x * blockDim.x + threadIdx.x; i < E;
         i += gridDim.x * blockDim.x) {
        float w = ew[i];
        mn = fminf(mn, w);
        mx = fmaxf(mx, w);
    }
    for (int off = 16; off > 0; off >>= 1) {
        mn = fminf(mn, __shfl_down(mn, off, 32));
        mx = fmaxf(mx, __shfl_down(mx, off, 32));
    }
    if ((threadIdx.x & 31) == 0) {
        atomicMin(&mm[0], __float_as_uint(mn));
        atomicMax(&mm[1], __float_as_uint(mx));
    }
}

// ---------------------------------------------------------------------------
// per-destination degree count
// ---------------------------------------------------------------------------
__global__ void count_kernel(const int* __restrict__ dst, float* __restrict__ cnt,
                             int E) {
    int e = blockIdx.x * blockDim.x + threadIdx.x;
    if (e < E) atomicAdd(&cnt[dst[e]], 1.0f);
}

__global__ void invcnt_kernel(const float* __restrict__ cnt,
                              float* __restrict__ inv, int N) {
    int n = blockIdx.x * blockDim.x + threadIdx.x;
    if (n < N) inv[n] = 1.0f / fmaxf(cnt[n], 1.0f);
}

// ---------------------------------------------------------------------------
// WMMA f32 GEMM:  C[N x M] = A[N x 128] * B[128 x M]  (row-major, M in {64,128})
// One wave per 16-row tile. A fragments for full K=128 in 64 VGPRs.
// Per column tile: batch-load all 32 B fragments (64 VGPRs), then 32
// back-to-back v_wmma_f32_16x16x4_f32 on two interleaved accumulators.
// M is a template parameter so all load/store offsets are immediates.
// EXEC is all ones (no divergence; N % 16 == 0).
//
// Fragment layouts per CDNA5 ISA 7.12.2:
//   A 16x4 f32 : lanes 0-15 -> rows, VGPR j holds k = j + 2*(lane>=16)
//   B 4x16 f32 : mirrored: VGPR j, lane l holds B[4s + j + 2*(l>=16)][l%16]
//   C 16x16 f32: VGPR j, lanes 0-15 -> row j, lanes 16-31 -> row j+8
// ---------------------------------------------------------------------------
template <int M>
__global__ __launch_bounds__(32)
void gemm_wmma_f32_kernel(const float* __restrict__ A, const float* __restrict__ B,
                          float* __restrict__ C) {
    const int K = 128;
    const int lane = threadIdx.x;
    const int half = lane >> 4;   // 0|1
    const int lr   = lane & 15;   // 0..15
    const int rowbase = blockIdx.x * 16;

    // ---- A fragments: base + compile-time offsets ----
    const float* arow = A + (size_t)(rowbase + lr) * K + 2 * half;
    v2f a[32];
#pragma unroll
    for (int s = 0; s < 32; ++s) {
        a[s] = *(const v2f*)(arow + 4 * s);
    }

#pragma unroll
    for (int ct = 0; ct < (M >> 4); ++ct) {
        const int col = ct * 16 + lr;
        // fold runtime 2*half*M into base; (4s)*M, (4s+1)*M are immediates
        const float* bp = B + (size_t)(2 * half) * M + col;
        v2f b[32];
#pragma unroll
        for (int s = 0; s < 32; ++s) {
            b[s].x = bp[(4 * s + 0) * M];
            b[s].y = bp[(4 * s + 1) * M];
        }
        v8f acc0 = {};
        v8f acc1 = {};
#pragma unroll
        for (int s = 0; s < 32; s += 2) {
            acc0 = __builtin_amdgcn_wmma_f32_16x16x4_f32(
                false, a[s], false, b[s], (short)0, acc0, false, false);
            acc1 = __builtin_amdgcn_wmma_f32_16x16x4_f32(
                false, a[s + 1], false, b[s + 1], (short)0, acc1, false, false);
        }
        float* crow = C + (size_t)(rowbase + 8 * half) * M + col;
#pragma unroll
        for (int j = 0; j < 8; ++j) {
            crow[(size_t)j * M] = acc0[j] + acc1[j];
        }
    }
}

// ---------------------------------------------------------------------------
// scatter for one relation, 4 columns per thread:
//   agg[dst[e], c..c+3] += proj[src[e], c..c+3] * w_norm(e)
// ---------------------------------------------------------------------------
__global__ void scatter_rel_kernel(const float* __restrict__ proj,
                                   const int* __restrict__ src,
                                   const int* __restrict__ dst,
                                   const int* __restrict__ et,
                                   const float* __restrict__ ew,
                                   const unsigned* __restrict__ mm,
                                   float* __restrict__ agg,
                                   int E, int M, int r) {
    const int Mq = M >> 2;
    long long idx = (long long)blockIdx.x * blockDim.x + threadIdx.x;
    long long total = (long long)E * Mq;
    if (idx >= total) return;
    int e  = (int)(idx / Mq);
    int c4 = (int)(idx - (long long)e * Mq) << 2;
    if (et[e] != r) return;
    float mn = __uint_as_float(mm[0]);
    float mx = __uint_as_float(mm[1]);
    float w = (ew[e] - mn) / (mx - mn + 1e-8f);
    const float4 p = *(const float4*)(proj + (size_t)src[e] * M + c4);
    float* ag = agg + (size_t)dst[e] * M + c4;
    atomicAdd(ag + 0, p.x * w);
    atomicAdd(ag + 1, p.y * w);
    atomicAdd(ag + 2, p.z * w);
    atomicAdd(ag + 3, p.w * w);
}

// ---------------------------------------------------------------------------
// out[n,c] = (agg[n,c]*inv[n] + out[n,c] + bias[c]), optional relu
// (out already holds the root transform x @ root)
// ---------------------------------------------------------------------------
__global__ void finalize_kernel(const float* __restrict__ agg,
                                const float* __restrict__ inv,
                                const float* __restrict__ bias,
                                float* __restrict__ out,
                                int N, int M, int do_relu) {
    long long idx = (long long)blockIdx.x * blockDim.x + threadIdx.x;
    if (idx >= (long long)N * M) return;
    int n = (int)(idx / M);
    int c = (int)(idx - (long long)n * M);
    float v = agg[idx] * inv[n] + out[idx] + bias[c];
    out[idx] = do_relu ? fmaxf(v, 0.0f) : v;
}

// ---------------------------------------------------------------------------
extern "C" void kernel_launch(void* const* d_in, const int* in_sizes, int n_in,
                              void* d_out, int out_size, void* d_ws, size_t ws_size,
                              hipStream_t stream) {
    const float* x     = (const float*)d_in[0];
    const int*   eidx  = (const int*)d_in[1];     // [2, E]
    const float* ew    = (const float*)d_in[2];
    const int*   et    = (const int*)d_in[3];
    const float* W1    = (const float*)d_in[4];   // [R, F, H]
    const float* root1 = (const float*)d_in[5];   // [F, H]
    const float* bias1 = (const float*)d_in[6];   // [H]
    const float* W2    = (const float*)d_in[7];   // [R, H, O]
    const float* root2 = (const float*)d_in[8];   // [H, O]
    const float* bias2 = (const float*)d_in[9];   // [O]
    float* out = (float*)d_out;                   // [N, O]

    const int* src = eidx;
    const int* dst = eidx + EE;

    // workspace carve-up (floats)
    float* proj = (float*)d_ws;                 // N*H
    float* agg  = proj + (size_t)NN * HH;       // N*H
    float* h    = agg  + (size_t)NN * HH;       // N*H
    float* cnt  = h    + (size_t)NN * HH;       // N
    float* inv  = cnt  + NN;                    // N
    unsigned* mm = (unsigned*)(inv + NN);       // 2

    const int T = 256;
    const int rowTiles = NN / 16;  // 3125

    // ---- shared prep ----
    hipMemsetAsync(agg, 0, (size_t)NN * HH * sizeof(float), stream);
    hipMemsetAsync(cnt, 0, (size_t)NN * sizeof(float), stream);
    init_mm_kernel<<<1, 32, 0, stream>>>(mm);
    ew_minmax_kernel<<<1024, T, 0, stream>>>(ew, EE, mm);
    count_kernel<<<(EE + T - 1) / T, T, 0, stream>>>(dst, cnt, EE);
    invcnt_kernel<<<(NN + T - 1) / T, T, 0, stream>>>(cnt, inv, NN);

    // ---- layer 1: H outputs, relu ----
    {
        long long sw = (long long)EE * (HH / 4);
        int sblocks = (int)((sw + T - 1) / T);
        for (int r = 0; r < RR; ++r) {
            gemm_wmma_f32_kernel<HH><<<rowTiles, 32, 0, stream>>>(
                x, W1 + (size_t)r * FF * HH, proj);
            scatter_rel_kernel<<<sblocks, T, 0, stream>>>(
                proj, src, dst, et, ew, mm, agg, EE, HH, r);
        }
        gemm_wmma_f32_kernel<HH><<<rowTiles, 32, 0, stream>>>(x, root1, h);
        long long fw = (long long)NN * HH;
        finalize_kernel<<<(int)((fw + T - 1) / T), T, 0, stream>>>(
            agg, inv, bias1, h, NN, HH, 1);
    }

    // ---- layer 2: O outputs, no relu ----
    {
        hipMemsetAsync(agg, 0, (size_t)NN * OO * sizeof(float), stream);
        long long sw = (long long)EE * (OO / 4);
        int sblocks = (int)((sw + T - 1) / T);
        for (int r = 0; r < RR; ++r) {
            gemm_wmma_f32_kernel<OO><<<rowTiles, 32, 0, stream>>>(
                h, W2 + (size_t)r * HH * OO, proj);
            scatter_rel_kernel<<<sblocks, T, 0, stream>>>(
                proj, src, dst, et, ew, mm, agg, EE, OO, r);
        }
        gemm_wmma_f32_kernel<OO><<<rowTiles, 32, 0, stream>>>(h, root2, out);
        long long fw = (long long)NN * OO;
        finalize_kernel<<<(int)((fw + T - 1) / T), T, 0, stream>>>(
            agg, inv, bias2, out, NN, OO, 0);
    }
}